// Transformer_GNN_18674517803415
// MI455X (gfx1250) — compile-verified
//
#include <hip/hip_runtime.h>
#include <math.h>

// Problem constants (match reference)
constexpr int NN = 50000;   // nodes
constexpr int EE = 800000;  // edges
constexpr int FF = 128;     // feature dim
constexpr int HH = 8;       // heads
constexpr int DD = 16;      // head dim
constexpr int LL = 4;       // layers

typedef float v2f __attribute__((ext_vector_type(2)));
typedef float v8f __attribute__((ext_vector_type(8)));

// Float atomic max via sign-split int/uint monotonic trick (init must be -inf).
__device__ __forceinline__ void atomicMaxF32(float* addr, float val) {
  if (val >= 0.0f) atomicMax((int*)addr, __float_as_int(val));
  else             atomicMin((unsigned int*)addr, __float_as_uint(val));
}

// ---------------------------------------------------------------------------
// Per-layer stat reset: amax = -inf, denom = 0
// ---------------------------------------------------------------------------
__global__ void init_stats(float* __restrict__ amax, float* __restrict__ denom, int n) {
  int i = blockIdx.x * blockDim.x + threadIdx.x;
  if (i < n) { amax[i] = -INFINITY; denom[i] = 0.0f; }
}

// ---------------------------------------------------------------------------
// Fused Q/K/V/Skip projection GEMM, register-blocked over M.
// V_WMMA_F32_16X16X4_F32 (full fp32). For a fixed (mat, tN) a lane's whole
// K=128 B-fragment is 32 x v2f -> loaded ONCE into VGPRs, then reused across
// 25 M-tiles. Steady state: 1 VMEM (A b64) per WMMA instead of 3.
// Grid: 4 mats * 8 Ntiles * 125 groups = 4000 waves = 500 blocks of 256.
// ---------------------------------------------------------------------------
constexpr int MT = 25;       // M-tiles per wave (3125 = 125 * 25)

__global__ void gemm_qkvs(const float* __restrict__ in,
    const float* __restrict__ Wq, const float* __restrict__ bq,
    const float* __restrict__ Wk, const float* __restrict__ bk,
    const float* __restrict__ Wv, const float* __restrict__ bv,
    const float* __restrict__ Ws, const float* __restrict__ bs,
    float* __restrict__ Q, float* __restrict__ K,
    float* __restrict__ V, float* __restrict__ S)
{
  const int lane = threadIdx.x & 31;
  const int wid  = (blockIdx.x * blockDim.x + threadIdx.x) >> 5;  // 0..3999
  const int mat  = wid / 1000;          // 0..3
  const int rem  = wid % 1000;
  const int tN   = rem / 125;           // 0..7
  const int grp  = rem % 125;           // 0..124
  const int tM0  = grp * MT;

  const float* W; const float* bias; float* out;
  if      (mat == 0) { W = Wq; bias = bq; out = Q; }
  else if (mat == 1) { W = Wk; bias = bk; out = K; }
  else if (mat == 2) { W = Wv; bias = bv; out = V; }
  else               { W = Ws; bias = bs; out = S; }

  const int hi  = lane >> 4;            // 0: K pair {k,k+1}; 1: {k+2,k+3}
  const int m   = lane & 15;
  const int col = tN * 16 + m;
  const float bb = bias[col];

  // B fragment for the whole K dimension, loaded once per wave.
  v2f breg[32];
  #pragma unroll
  for (int i = 0; i < 32; ++i) {
    const int ka = i * 4 + hi * 2;
    breg[i][0] = W[(size_t)ka       * FF + col];   // B[ka][col]
    breg[i][1] = W[(size_t)(ka + 1) * FF + col];   // B[ka+1][col]
  }

  for (int t = 0; t < MT; ++t) {
    const int tM = tM0 + t;
    const float2* arow = (const float2*)(in + (size_t)(tM * 16 + m) * FF);
    v8f acc = {};
    #pragma unroll
    for (int i = 0; i < 32; ++i) {
      float2 a2 = arow[(i * 4 + hi * 2) >> 1];     // A[m][ka], A[m][ka+1]
      v2f a; a[0] = a2.x; a[1] = a2.y;
      acc = __builtin_amdgcn_wmma_f32_16x16x4_f32(
          /*neg_a=*/false, a, /*neg_b=*/false, breg[i],
          /*c_mod=*/(short)0, acc, /*reuse_a=*/false, /*reuse_b=*/false);
    }
    const int rbase = tM * 16 + hi * 8;            // C/D: lane<16 -> M=j, else M=8+j
    #pragma unroll
    for (int j = 0; j < 8; ++j)
      out[(size_t)(rbase + j) * FF + col] = acc[j] + bb;
  }
}

// ---------------------------------------------------------------------------
// Pass 1 over edges: alpha logits + segment max.  One thread per (edge, head).
// ---------------------------------------------------------------------------
__global__ void edge_alpha(const int* __restrict__ src, const int* __restrict__ dst,
                           const float* __restrict__ Q, const float* __restrict__ K,
                           float* __restrict__ alph, float* __restrict__ amax)
{
  int idx = blockIdx.x * blockDim.x + threadIdx.x;   // e*H + h
  int e = idx >> 3, h = idx & 7;
  int s = src[e], d = dst[e];
  const float4* qp = (const float4*)(Q + (size_t)d * FF + h * DD);
  const float4* kp = (const float4*)(K + (size_t)s * FF + h * DD);
  float dot = 0.0f;
  #pragma unroll
  for (int i = 0; i < 4; ++i) {
    float4 q4 = qp[i], k4 = kp[i];
    dot += q4.x * k4.x + q4.y * k4.y + q4.z * k4.z + q4.w * k4.w;
  }
  dot *= 0.25f;                                      // 1/sqrt(16)
  alph[idx] = dot;                                   // logits into output region
  atomicMaxF32(&amax[d * HH + h], dot);
}

// ---------------------------------------------------------------------------
// Pass 2: p = exp(alpha - amax[dst]); denom += p.  In-place in alpha region.
// ---------------------------------------------------------------------------
__global__ void edge_exp(const int* __restrict__ dst,
                         float* __restrict__ alph,
                         const float* __restrict__ amax, float* __restrict__ denom)
{
  int idx = blockIdx.x * blockDim.x + threadIdx.x;
  int e = idx >> 3, h = idx & 7;
  int d = dst[e];
  float p = __expf(alph[idx] - amax[d * HH + h]);
  alph[idx] = p;
  atomicAdd(&denom[d * HH + h], p);
}

// ---------------------------------------------------------------------------
// Pass 3: normalize attention, write it back (final output value), and
// scatter a * v[src] into the destination accumulator (pre-loaded with skip).
// One wave32 per edge; lane handles 4 contiguous features of head lane/4.
// ---------------------------------------------------------------------------
__global__ void edge_scatter(const int* __restrict__ src, const int* __restrict__ dst,
                             const float* __restrict__ V, float* __restrict__ alph,
                             const float* __restrict__ denom, float* __restrict__ acc)
{
  int gid  = blockIdx.x * blockDim.x + threadIdx.x;
  int e    = gid >> 5;
  int lane = gid & 31;
  int s = src[e], d = dst[e];
  int h = lane >> 2;
  float a = alph[(size_t)e * HH + h] / denom[d * HH + h];
  if ((lane & 3) == 0) alph[(size_t)e * HH + h] = a;   // finalize attention output
  int f0 = lane * 4;
  float4 v4 = *(const float4*)(V + (size_t)s * FF + f0);
  float* ap = acc + (size_t)d * FF + f0;
  atomicAdd(ap + 0, a * v4.x);
  atomicAdd(ap + 1, a * v4.y);
  atomicAdd(ap + 2, a * v4.z);
  atomicAdd(ap + 3, a * v4.w);
}

// ---------------------------------------------------------------------------
// Exact GELU (erf form), in place, between layers.
// ---------------------------------------------------------------------------
__global__ void gelu_inplace(float* __restrict__ h) {
  int i = blockIdx.x * blockDim.x + threadIdx.x;
  float x = h[i];
  h[i] = 0.5f * x * (1.0f + erff(x * 0.70710678118654752f));
}

// ---------------------------------------------------------------------------
extern "C" void kernel_launch(void* const* d_in, const int* in_sizes, int n_in,
                              void* d_out, int out_size, void* d_ws, size_t ws_size,
                              hipStream_t stream)
{
  (void)in_sizes; (void)n_in; (void)out_size; (void)ws_size;

  const float* x  = (const float*)d_in[0];
  const int*   ei = (const int*)d_in[1];     // [2,E]: row 0 = src, row 1 = dst
  const int*   src = ei;
  const int*   dst = ei + EE;
  const float* Wq = (const float*)d_in[2];
  const float* bq = (const float*)d_in[3];
  const float* Wk = (const float*)d_in[4];
  const float* bk = (const float*)d_in[5];
  const float* Wv = (const float*)d_in[6];
  const float* bv = (const float*)d_in[7];
  const float* Ws = (const float*)d_in[8];
  const float* bs = (const float*)d_in[9];

  float* out_h  = (float*)d_out;                    // [N, F]
  float* out_al = out_h + (size_t)NN * FF;          // [L, E, H]

  // Workspace layout (~131 MB): hA, hB, Q, K, V, amax, denom
  const size_t NF = (size_t)NN * FF;
  float* ws    = (float*)d_ws;
  float* bufA  = ws;
  float* bufB  = ws + NF;
  float* Qb    = ws + 2 * NF;
  float* Kb    = ws + 3 * NF;
  float* Vb    = ws + 4 * NF;
  float* amax  = ws + 5 * NF;
  float* denom = amax + (size_t)NN * HH;

  const float* cur = x;
  for (int l = 0; l < LL; ++l) {
    float* nxt  = (l == LL - 1) ? out_h : ((l & 1) ? bufB : bufA);
    float* alph = out_al + (size_t)l * EE * HH;

    init_stats<<<(NN * HH + 255) / 256, 256, 0, stream>>>(amax, denom, NN * HH);

    // 4000 waves -> 500 blocks of 256 (8 waves)
    gemm_qkvs<<<500, 256, 0, stream>>>(cur,
        Wq + (size_t)l * FF * FF, bq + (size_t)l * FF,
        Wk + (size_t)l * FF * FF, bk + (size_t)l * FF,
        Wv + (size_t)l * FF * FF, bv + (size_t)l * FF,
        Ws + (size_t)l * FF * FF, bs + (size_t)l * FF,
        Qb, Kb, Vb, nxt);

    edge_alpha  <<<EE * HH / 256,  256, 0, stream>>>(src, dst, Qb, Kb, alph, amax);
    edge_exp    <<<EE * HH / 256,  256, 0, stream>>>(dst, alph, amax, denom);
    edge_scatter<<<EE * 32 / 256,  256, 0, stream>>>(src, dst, Vb, alph, denom, nxt);

    if (l < LL - 1)
      gelu_inplace<<<(int)(NF / 256), 256, 0, stream>>>(nxt);

    cur = nxt;
  }
}